// HeteroMLPPredictor_23605140259308
// MI455X (gfx1250) — compile-verified
//
#include <hip/hip_runtime.h>

#define N_NODES 100000
#define N_EDGES 100000
#define DD      256
#define N_TILES (N_EDGES / 16)   // 6250

typedef float v2f __attribute__((ext_vector_type(2)));
typedef float v8f __attribute__((ext_vector_type(8)));

// Kernel 1: build side matrices  M_s[f,d] = d_side[s,f] * R_W[f,d] * d_side[s,d]
__global__ void prep_side_kernel(const float* __restrict__ R_W,
                                 const float* __restrict__ d_side,
                                 float* __restrict__ Mside) {
    int idx = blockIdx.x * blockDim.x + threadIdx.x;   // 0 .. 2*256*256-1
    int s = idx >> 16;
    int f = (idx >> 8) & 255;
    int d = idx & 255;
    Mside[idx] = d_side[s * DD + f] * R_W[f * DD + d] * d_side[s * DD + d];
}

// Kernel 2: one wave per 16-edge tile of one relation.
//   T = HU * M^T via f32 WMMA (A register-resident, B streamed from L2/WGP$),
//   score[e] = sum_f (T[e,f] + b[f]) * HV[e,f]
__global__ __launch_bounds__(64) void score_kernel(
    const float* __restrict__ h,
    const int*   __restrict__ src_idx,
    const int*   __restrict__ dst_idx,
    const float* __restrict__ Mside,
    const float* __restrict__ W_rel,
    const float* __restrict__ b_rel,
    float*       __restrict__ out) {

    const int lane = threadIdx.x & 31;
    const int wave = threadIdx.x >> 5;
    const int lm   = lane & 15;      // matrix row/col within half
    const int hi   = lane >> 4;      // which 16-lane half
    const int c    = blockIdx.y;     // relation 0..3
    const int tile = blockIdx.x * 2 + wave;
    const int e0   = tile * 16;
    const int base = c * N_EDGES + e0;

    const float* Mmat = (c < 2) ? (Mside + c * DD * DD)
                                : (W_rel + (c - 2) * DD * DD);
    const bool   has_b = (c >= 2);
    const float* brow  = has_b ? (b_rel + (c - 2) * DD) : b_rel; // dummy if unused

    // ---- Gather A operand (HU tile, 16 edges x 256) into registers ----
    // A-layout (16x4 f32): lane = m + 16*(k>=2), vgpr j holds k = 2*hi + j.
    const int su = src_idx[base + lm];
    const float* pu = h + (long)su * DD;
    v2f a[64];
#pragma unroll
    for (int kk = 0; kk < 64; ++kk)
        a[kk] = *(const v2f*)(pu + kk * 4 + hi * 2);

    // ---- Hoisted dst-row offsets: lane needs row of edge (j + 8*hi) ----
    const int hvofs = dst_idx[base + lm] * DD;
    int ofs[8];
#pragma unroll
    for (int j = 0; j < 8; ++j)
        ofs[j] = __shfl(hvofs, j + 8 * hi, 32);

    float partial[8];
#pragma unroll
    for (int j = 0; j < 8; ++j) partial[j] = 0.0f;

    // ---- Loop over 16 output f-tiles; 64 WMMA (K=256) each, 2 acc chains ----
    for (int ft = 0; ft < 16; ++ft) {
        // B[k,n] = M[ft*16+n, kbase+k]; lane = n + 16*(k>=2), vgpr j -> k=2*hi+j
        const float* pb = Mmat + (ft * 16 + lm) * DD;

        v8f acc0 = {0.f, 0.f, 0.f, 0.f, 0.f, 0.f, 0.f, 0.f};
        v8f acc1 = {0.f, 0.f, 0.f, 0.f, 0.f, 0.f, 0.f, 0.f};
#pragma unroll
        for (int kk = 0; kk < 64; kk += 2) {
            v2f b0 = *(const v2f*)(pb + kk * 4 + hi * 2);
            v2f b1 = *(const v2f*)(pb + (kk + 1) * 4 + hi * 2);
            acc0 = __builtin_amdgcn_wmma_f32_16x16x4_f32(
                false, a[kk], false, b0, (short)0, acc0, false, false);
            acc1 = __builtin_amdgcn_wmma_f32_16x16x4_f32(
                false, a[kk + 1], false, b1, (short)0, acc1, false, false);
        }

        const float bfrag = has_b ? brow[ft * 16 + lm] : 0.0f;

        // C/D layout: acc[j] at lane l = T[e = j + 8*hi, f = ft*16 + (l&15)]
#pragma unroll
        for (int j = 0; j < 8; ++j) {
            const float hvv = h[ofs[j] + ft * 16 + lm];
            partial[j] += (acc0[j] + acc1[j] + bfrag) * hvv;
        }
    }

    // ---- Reduce each partial across its 16-lane half and store ----
#pragma unroll
    for (int j = 0; j < 8; ++j) {
        float v = partial[j];
        v += __shfl_xor(v, 1, 32);
        v += __shfl_xor(v, 2, 32);
        v += __shfl_xor(v, 4, 32);
        v += __shfl_xor(v, 8, 32);
        if (lm == 0)
            out[base + j + 8 * hi] = v;
    }
}

extern "C" void kernel_launch(void* const* d_in, const int* in_sizes, int n_in,
                              void* d_out, int out_size, void* d_ws, size_t ws_size,
                              hipStream_t stream) {
    const float* h       = (const float*)d_in[0];
    const int*   src_idx = (const int*)  d_in[1];
    const int*   dst_idx = (const int*)  d_in[2];
    const float* R_W     = (const float*)d_in[3];
    const float* d_side  = (const float*)d_in[4];
    const float* W_rel   = (const float*)d_in[5];
    const float* b_rel   = (const float*)d_in[6];
    float* out   = (float*)d_out;
    float* Mside = (float*)d_ws;      // 2 * 256 * 256 floats = 512 KB

    // Build effective side matrices.
    prep_side_kernel<<<(2 * DD * DD) / 256, 256, 0, stream>>>(R_W, d_side, Mside);

    // 6250 tiles per relation, 2 waves (tiles) per 64-thread block -> 3125 blocks.
    dim3 grid(N_TILES / 2, 4);
    score_kernel<<<grid, 64, 0, stream>>>(h, src_idx, dst_idx, Mside, W_rel, b_rel, out);
}